// FindCorrespondence_32547262169638
// MI455X (gfx1250) — compile-verified
//
#include <hip/hip_runtime.h>
#include <stdint.h>

// Problem constants (reference: B=4, H=W=64, HW=4096, BETA=50, SIGMA=5)
#define NB 4
#define HH 64
#define WW 64
#define HW 4096

// BETA * log2(e)  (softmax exp via v_exp_f32)
#define K_BETA_LOG2E 72.13475204444817f
// log2(e) / (2*SIGMA^2)  (gaussian via exp2)
#define C_GAUSS_LOG2E 0.028853900817779268f

#if __has_builtin(__builtin_amdgcn_s_wait_asynccnt)
#define WAIT_ASYNC(n) __builtin_amdgcn_s_wait_asynccnt(n)
#else
#define WAIT_ASYNC(n) asm volatile("s_wait_asynccnt " #n ::: "memory")
#endif

__device__ __forceinline__ void f4arr(const float4 v, float* a) {
    a[0] = v.x; a[1] = v.y; a[2] = v.z; a[3] = v.w;
}

// One block per (b, th): 256 threads = 8 waves.
// Thread = (twq = tid&15 -> tw quad twq*4..+3, r = tid>>4 -> source chunk / tile row group)
__global__ __launch_bounds__(256) void
FindCorrespondence_32547262169638_kernel(const float* __restrict__ corr,
                                         float* __restrict__ out)
{
    const int bt  = blockIdx.x;      // 0..255
    const int b   = bt >> 6;
    const int th  = bt & 63;
    const int tid = threadIdx.x;
    const int twq = tid & 15;        // tw quad
    const int r   = tid >> 4;        // 0..15

    // ---- LDS ----
    __shared__ float lutx[64][64];          // [sx][tw]  16 KB
    __shared__ float luty[64][64];          // [sy][tw]  16 KB
    __shared__ float tileBuf[2][32][64];    // [buf][s_sub][tw] 16 KB (per-thread staging, no sharing)
    __shared__ union {
        struct { float v[16][64]; unsigned idx[16][64]; } s1;   // 8 KB (argmax)
        struct { float s0[16][64]; float sx[16][64]; float sy[16][64]; } s2; // 12 KB
    } red;
    __shared__ float sKM[64];               // K * m per tw
    __shared__ float sIdxX[64], sIdxY[64];

    // element (s, tw) of this block's slice lives at base[s*4096 + tw]
    const float* base = corr + (size_t)b * ((size_t)HW * HW) + (size_t)th * WW;

    // =========================================================
    // Sweep 1: argmax (and max) over source dim, coalesced b128
    // =========================================================
    float    bv[4] = {-3.402823466e38f, -3.402823466e38f, -3.402823466e38f, -3.402823466e38f};
    unsigned bi[4] = {0u, 0u, 0u, 0u};

    #pragma unroll 4
    for (int i = 0; i < 256; ++i) {
        const int s = r * 256 + i;
        const float* p = base + (size_t)s * HW + twq * 4;
        __builtin_prefetch(p + (size_t)32 * HW, 0, 1);   // global_prefetch_b8, stream ahead into GL2
        float c[4]; f4arr(*reinterpret_cast<const float4*>(p), c);
        #pragma unroll
        for (int j = 0; j < 4; ++j) {
            if (c[j] > bv[j]) { bv[j] = c[j]; bi[j] = (unsigned)s; }  // strict > : first max wins
        }
    }
    #pragma unroll
    for (int j = 0; j < 4; ++j) {
        red.s1.v[r][twq * 4 + j]   = bv[j];
        red.s1.idx[r][twq * 4 + j] = bi[j];
    }
    __syncthreads();

    if (tid < 64) {
        float bestv = red.s1.v[0][tid];
        unsigned besti = red.s1.idx[0][tid];
        #pragma unroll
        for (int k = 1; k < 16; ++k) {
            float v = red.s1.v[k][tid];
            unsigned ii = red.s1.idx[k][tid];
            if (v > bestv || (v == bestv && ii < besti)) { bestv = v; besti = ii; }
        }
        sIdxX[tid] = (float)(besti & 63u);
        sIdxY[tid] = (float)(besti >> 6);
        // softmax max: gaussian is exactly 1 at the argmax and g*corr <= max(corr_max, 0)
        sKM[tid] = K_BETA_LOG2E * fmaxf(bestv, 0.0f);
    }
    __syncthreads();

    // =========================================================
    // Build factored gaussian LUTs (transposed: [coord][tw])
    // =========================================================
    for (int k = tid; k < 4096; k += 256) {
        const int tw = k & 63, c = k >> 6;
        const float dx = (float)c - sIdxX[tw];
        const float dy = (float)c - sIdxY[tw];
        lutx[c][tw] = exp2f(-C_GAUSS_LOG2E * dx * dx);
        luty[c][tw] = exp2f(-C_GAUSS_LOG2E * dy * dy);
    }
    __syncthreads();

    // =========================================================
    // Sweep 2: softmax + soft-argmax.
    // Streamed via double-buffered global_load_async_to_lds_b128.
    // Each thread consumes ONLY the LDS words it loaded itself, so
    // per-wave s_wait_asynccnt suffices — NO workgroup barriers in
    // the hot loop.
    // =========================================================
    float a0[4] = {0.f, 0.f, 0.f, 0.f};
    float ax[4] = {0.f, 0.f, 0.f, 0.f};
    float ay[4] = {0.f, 0.f, 0.f, 0.f};
    float km4[4];
    #pragma unroll
    for (int j = 0; j < 4; ++j) km4[j] = sKM[twq * 4 + j];

    auto issueTile = [&](int t, int buf) {
        #pragma unroll
        for (int q = 0; q < 2; ++q) {
            const int row = r + q * 16;                  // 0..31
            const int s   = t * 32 + row;
            const float* gp = base + (size_t)s * HW + twq * 4;
            const uint32_t lp = (uint32_t)(uintptr_t)&tileBuf[buf][row][twq * 4];
            asm volatile("global_load_async_to_lds_b128 %0, %1, off"
                         :: "v"(lp), "v"((uint64_t)(uintptr_t)gp)
                         : "memory");
        }
    };

    issueTile(0, 0);
    for (int t = 0; t < 128; ++t) {
        const int buf = t & 1;
        if (t + 1 < 128) {
            issueTile(t + 1, buf ^ 1);   // keep next tile in flight
            WAIT_ASYNC(2);               // this wave's tile-t loads complete (in-order)
        } else {
            WAIT_ASYNC(0);
        }

        // sy is constant across a 32-row tile: hoist luty + yn
        const int   sy = t >> 1;
        const float yn = fmaf((float)sy, 2.0f / 63.0f, -1.0f);
        float ly[4];
        f4arr(*reinterpret_cast<const float4*>(&luty[sy][twq * 4]), ly);

        #pragma unroll
        for (int q = 0; q < 2; ++q) {
            const int row = r + q * 16;
            const int sx  = ((t & 1) << 5) + row;        // (32t + row) & 63
            const float xn = fmaf((float)sx, 2.0f / 63.0f, -1.0f);

            float c[4], lx[4];
            f4arr(*reinterpret_cast<const float4*>(&tileBuf[buf][row][twq * 4]), c);
            f4arr(*reinterpret_cast<const float4*>(&lutx[sx][twq * 4]), lx);

            #pragma unroll
            for (int j = 0; j < 4; ++j) {
                const float v = c[j] * (lx[j] * ly[j]);
                const float e = exp2f(fmaf(K_BETA_LOG2E, v, -km4[j]));   // v_exp_f32
                a0[j] += e;
                ax[j] = fmaf(e, xn, ax[j]);
                ay[j] = fmaf(e, yn, ay[j]);
            }
        }
    }

    // ---- cross-chunk reduction + output ----
    #pragma unroll
    for (int j = 0; j < 4; ++j) {
        red.s2.s0[r][twq * 4 + j] = a0[j];
        red.s2.sx[r][twq * 4 + j] = ax[j];
        red.s2.sy[r][twq * 4 + j] = ay[j];
    }
    __syncthreads();

    if (tid < 64) {
        float S = 0.f, X = 0.f, Y = 0.f;
        #pragma unroll
        for (int k = 0; k < 16; ++k) {
            S += red.s2.s0[k][tid];
            X += red.s2.sx[k][tid];
            Y += red.s2.sy[k][tid];
        }
        const float inv = 1.0f / S;      // S >= 1 (argmax term contributes exp2(0)=1)
        const size_t o = (((size_t)b * HH + th) * WW + tid) * 2;
        out[o + 0] = X * inv;            // grid[..., 0] = gx
        out[o + 1] = Y * inv;            // grid[..., 1] = gy
        // flow = zeros, layout [b, 2, h, w], offset after grid (32768 floats)
        const size_t fo = 32768 + ((size_t)b * 2) * (size_t)HW + (size_t)th * WW + tid;
        out[fo]      = 0.0f;
        out[fo + HW] = 0.0f;
    }
}

extern "C" void kernel_launch(void* const* d_in, const int* in_sizes, int n_in,
                              void* d_out, int out_size, void* d_ws, size_t ws_size,
                              hipStream_t stream) {
    (void)in_sizes; (void)n_in; (void)d_ws; (void)ws_size; (void)out_size;
    const float* corr = (const float*)d_in[0];
    float* out = (float*)d_out;
    // one block per (b, th): 4 * 64 = 256 blocks, 256 threads each
    FindCorrespondence_32547262169638_kernel<<<dim3(NB * HH), dim3(256), 0, stream>>>(corr, out);
}